// Attention_59193239274210
// MI455X (gfx1250) — compile-verified
//
#include <hip/hip_runtime.h>

typedef __attribute__((ext_vector_type(2))) float v2f;
typedef __attribute__((ext_vector_type(4))) float v4f;
typedef __attribute__((ext_vector_type(8))) float v8f;
typedef __attribute__((ext_vector_type(4))) unsigned int v4u;
typedef __attribute__((ext_vector_type(4))) int v4i;
typedef __attribute__((ext_vector_type(8))) int v8i;

#define DD   1024
#define SS   1024
#define BB   8
#define HH   8
#define DHH  128
#define NEGVAL (-4294967295.0f)

// D = A(16x4, f32) * B(4x16, f32) + C(16x16, f32)
__device__ __forceinline__ v8f wmma4(v2f a, v2f b, v8f c) {
  return __builtin_amdgcn_wmma_f32_16x16x4_f32(
      /*neg_a=*/false, a, /*neg_b=*/false, b,
      /*c_mod=*/(short)0, c, /*reuse_a=*/false, /*reuse_b=*/false);
}

// ---------------------------------------------------------------------------
// Tensor Data Mover: issue a 2D tile load (rows x cols f32, global row stride
// in elements) into LDS at lds_byte_off, with LDS row padding so that the
// stored row stride is (cols + 4) dwords (bank-conflict-free fragment reads).
// pad_interval_code: 4 => pad every 32 dwords, 5 => 64, 6 => 128.
// pad_amount_code 3 => 4 dwords of pad.
// ---------------------------------------------------------------------------
__device__ __forceinline__ void tdm_load_2d(unsigned lds_byte_off,
                                            const float* gptr,
                                            unsigned tile_cols,
                                            unsigned tile_rows,
                                            unsigned long long row_stride_elems,
                                            unsigned pad_interval_code,
                                            unsigned pad_amount_code)
{
  unsigned long long ga = (unsigned long long)(uintptr_t)gptr;
  v4u g0;
  g0.x = 1u;                                             // count=1, user desc
  g0.y = lds_byte_off;                                   // lds_addr
  g0.z = (unsigned)(ga & 0xFFFFFFFFull);                 // global_addr[31:0]
  g0.w = (unsigned)((ga >> 32) & 0x01FFFFFFull) | (2u << 30);  // [56:32] | type=2

  unsigned flags = (2u << 16)                    // data_size = 4 bytes
                 | (1u << 20)                    // pad_enable
                 | (pad_interval_code << 22)
                 | (pad_amount_code << 25);
  unsigned long long td0 = tile_cols;            // tensor_dim0 (exact tile)
  unsigned long long td1 = tile_rows;            // tensor_dim1
  union { unsigned long long q[4]; v8i v; } g1;
  g1.q[0] = (unsigned long long)flags | (td0 << 48);                 // [79:48] dim0
  g1.q[1] = (td0 >> 16) | (td1 << 16) | ((unsigned long long)tile_cols << 48); // dim1,[127:112] tile_dim0
  g1.q[2] = (unsigned long long)tile_rows                             // [143:128] tile_dim1
          | ((row_stride_elems & 0xFFFFFFFFull) << 32);               // [207:160] stride0 (low)
  g1.q[3] = (row_stride_elems >> 32);                                 // stride0 (high), stride1=0

  v4i z4 = {0, 0, 0, 0};
#if defined(__clang_major__) && (__clang_major__ >= 23)
  v8i z8 = {0, 0, 0, 0, 0, 0, 0, 0};
  __builtin_amdgcn_tensor_load_to_lds(g0, g1.v, z4, z4, z8, 0);
#else
  __builtin_amdgcn_tensor_load_to_lds(g0, g1.v, z4, z4, 0);
#endif
}

// ---------------------------------------------------------------------------
// 64(M) x 128(N) tile GEMM core, TDM double-buffered. 256 threads = 8 waves
// (4M x 2N); wave 0 drives the Tensor Data Mover (prefetching the next K-slab
// while all waves run WMMA on the current one). A row-major (lda), B row-major
// (ldb), pre-offset to tile origin. As: 2 x 64x36 floats; Bs: 2 x 32x132.
// ---------------------------------------------------------------------------
#define AS_STRIDE 36
#define BS_STRIDE 132
#define AS_BUF (64 * AS_STRIDE)
#define BS_BUF (32 * BS_STRIDE)

__device__ __forceinline__ void gemm_64x128(const float* __restrict__ A, long lda,
                                            const float* __restrict__ B, long ldb,
                                            float* __restrict__ C, long ldc,
                                            int K, float* As, float* Bs)
{
  const int tid  = threadIdx.x;
  const int lane = tid & 31;
  const int wave = tid >> 5;
  const int wm   = wave & 3;            // M-wave: rows wm*16
  const int wn   = wave >> 2;           // N-wave: cols wn*64
  const int l16  = lane & 15;
  const int kh2  = (lane >> 4) << 1;    // 0 or 2 (K split across lane halves)
  const int h8   = (lane >> 4) << 3;    // C row offset for lanes 16..31
  const bool issuer = (wave == 0);
  const unsigned asOff = (unsigned)(uintptr_t)As;
  const unsigned bsOff = (unsigned)(uintptr_t)Bs;

  v8f acc[4];
#pragma unroll
  for (int i = 0; i < 4; ++i)
#pragma unroll
    for (int j = 0; j < 8; ++j) acc[i][j] = 0.0f;

  if (issuer) {                         // prologue: stage k0 = 0 into buffer 0
    tdm_load_2d(asOff, A, 32, 64, lda, 4, 3);
    tdm_load_2d(bsOff, B, 128, 32, ldb, 6, 3);
  }

  int buf = 0;
  for (int k0 = 0; k0 < K; k0 += 32, buf ^= 1) {
    if (issuer) {
      if (k0 + 32 < K) {                // prefetch next slab into other buffer
        tdm_load_2d(asOff + (unsigned)((buf ^ 1) * AS_BUF * 4),
                    A + k0 + 32, 32, 64, lda, 4, 3);
        tdm_load_2d(bsOff + (unsigned)((buf ^ 1) * BS_BUF * 4),
                    B + (long)(k0 + 32) * ldb, 128, 32, ldb, 6, 3);
        __builtin_amdgcn_s_wait_tensorcnt(2);  // current buffer's 2 DMAs done
      } else {
        __builtin_amdgcn_s_wait_tensorcnt(0);
      }
    }
    __syncthreads();                    // publish DMA'd LDS to all waves
    const float* Ab = As + buf * AS_BUF;
    const float* Bb = Bs + buf * BS_BUF;
#pragma unroll
    for (int kk = 0; kk < 32; kk += 4) {
      v2f a;
      a.x = Ab[(wm * 16 + l16) * AS_STRIDE + kk + kh2];
      a.y = Ab[(wm * 16 + l16) * AS_STRIDE + kk + kh2 + 1];
#pragma unroll
      for (int nt = 0; nt < 4; ++nt) {
        v2f b;
        b.x = Bb[(kk + kh2) * BS_STRIDE + wn * 64 + nt * 16 + l16];
        b.y = Bb[(kk + kh2 + 1) * BS_STRIDE + wn * 64 + nt * 16 + l16];
        acc[nt] = wmma4(a, b, acc[nt]);
      }
    }
    __syncthreads();                    // done reading buf before TDM reuses it
  }
#pragma unroll
  for (int nt = 0; nt < 4; ++nt)
#pragma unroll
    for (int j = 0; j < 8; ++j)
      C[(long)(wm * 16 + j + h8) * ldc + wn * 64 + nt * 16 + l16] = acc[nt][j];
}

// ----------------------- QKV projection: X @ W --------------------------
__global__ __launch_bounds__(256) void k_gemm_proj(const float* __restrict__ A,
                                                   const float* __restrict__ W,
                                                   float* __restrict__ Cout)
{
  __shared__ float As[2 * AS_BUF];
  __shared__ float Bs[2 * BS_BUF];
  long m0 = (long)blockIdx.y * 64;
  long n0 = (long)blockIdx.x * 128;
  gemm_64x128(A + m0 * DD, DD, W + n0, DD, Cout + m0 * DD + n0, DD, DD, As, Bs);
}

// ----------------------- attn @ V per (head,batch) ----------------------
__global__ __launch_bounds__(256) void k_gemm_av(const float* __restrict__ attn,
                                                 const float* __restrict__ Vb,
                                                 float* __restrict__ ctx)
{
  __shared__ float As[2 * AS_BUF];
  __shared__ float Bs[2 * BS_BUF];
  int hb = blockIdx.z, b = hb & 7, head = hb >> 3;
  long m0 = (long)blockIdx.y * 64;
  const float* Ab = attn + ((long)hb << 20) + m0 * SS;
  const float* Bb = Vb + ((long)b << 20) + head * DHH;
  float* Cb = ctx + ((long)b << 20) + m0 * DD + head * DHH;
  gemm_64x128(Ab, SS, Bb, DD, Cb, DD, SS, As, Bs);
}

// --------- scores = QK^T/sqrt(dh), mask, softmax, * query_mask ----------
__global__ __launch_bounds__(256) void k_attn(const float* __restrict__ Qb,
                                              const float* __restrict__ Kb,
                                              const unsigned char* __restrict__ mask,
                                              const float* __restrict__ qmask,
                                              float* __restrict__ attnOut)
{
  __shared__ float Qs[16 * 132];
  __shared__ float Ks[128 * 68];
  __shared__ float redbuf[16 * 8];
  const int tid = threadIdx.x, lane = tid & 31, wave = tid >> 5;
  const int l16 = lane & 15;
  const int kh2 = (lane >> 4) << 1;
  const int h8  = (lane >> 4) << 3;
  const int QS = 132, KS = 68;
  const int hb = blockIdx.y, b = hb & 7, head = hb >> 3;
  const long q0 = (long)blockIdx.x * 16;
  const long qrowbase = (long)b * SS + q0;

  // load Q tile 16x128
#pragma unroll
  for (int i = 0; i < 2; ++i) {
    int id = tid + i * 256;
    int r = id >> 5, c = (id & 31) << 2;
    v4f v = *(const v4f*)(Qb + (qrowbase + r) * DD + head * DHH + c);
    *(v4f*)(Qs + r * QS + c) = v;
  }

  v8f acc[8];
#pragma unroll
  for (int n = 0; n < 8; ++n)
#pragma unroll
    for (int j = 0; j < 8; ++j) acc[n][j] = 0.0f;

  for (int nb = 0; nb < 8; ++nb) {          // 128-column chunks of Sk
    for (int kh = 0; kh < 128; kh += 64) {  // dh halves (keeps LDS < 64KB)
      __syncthreads();
#pragma unroll
      for (int i = 0; i < 8; ++i) {
        int id = tid + i * 256;
        int n = id >> 4, c = (id & 15) << 2;
        v4f v = *(const v4f*)(Kb + ((long)b * SS + nb * 128 + n) * DD + head * DHH + kh + c);
        *(v4f*)(Ks + n * KS + c) = v;
      }
      __syncthreads();
#pragma unroll
      for (int kk = 0; kk < 64; kk += 4) {
        v2f a;
        a.x = Qs[l16 * QS + kh + kk + kh2];
        a.y = Qs[l16 * QS + kh + kk + kh2 + 1];
        v2f bb;  // B[k][n] = K[n][k]
        bb.x = Ks[(wave * 16 + l16) * KS + kk + kh2];
        bb.y = Ks[(wave * 16 + l16) * KS + kk + kh2 + 1];
        acc[nb] = wmma4(a, bb, acc[nb]);
      }
    }
  }

  const float scale = 0.08838834764831845f;  // 1/sqrt(128)
  float rmax[8], rsum[8];
#pragma unroll
  for (int j = 0; j < 8; ++j) rmax[j] = -3.4e38f;
#pragma unroll
  for (int nb = 0; nb < 8; ++nb) {
    int col = nb * 128 + wave * 16 + l16;
#pragma unroll
    for (int j = 0; j < 8; ++j) {
      int row = j + h8;
      float v = acc[nb][j] * scale;
      unsigned char mq = mask[(qrowbase + row) * (long)SS + col];
      v = mq ? NEGVAL : v;
      acc[nb][j] = v;
      rmax[j] = fmaxf(rmax[j], v);
    }
  }
#pragma unroll
  for (int off = 1; off < 16; off <<= 1)
#pragma unroll
    for (int j = 0; j < 8; ++j)
      rmax[j] = fmaxf(rmax[j], __shfl_xor(rmax[j], off, 32));
  if (l16 == 0) {
#pragma unroll
    for (int j = 0; j < 8; ++j) redbuf[(j + h8) * 8 + wave] = rmax[j];
  }
  __syncthreads();
  float rowmax[8];
#pragma unroll
  for (int j = 0; j < 8; ++j) {
    float m = -3.4e38f;
#pragma unroll
    for (int w = 0; w < 8; ++w) m = fmaxf(m, redbuf[(j + h8) * 8 + w]);
    rowmax[j] = m;
  }
  __syncthreads();

#pragma unroll
  for (int j = 0; j < 8; ++j) rsum[j] = 0.0f;
#pragma unroll
  for (int nb = 0; nb < 8; ++nb)
#pragma unroll
    for (int j = 0; j < 8; ++j) {
      float e = __expf(acc[nb][j] - rowmax[j]);
      acc[nb][j] = e;
      rsum[j] += e;
    }
#pragma unroll
  for (int off = 1; off < 16; off <<= 1)
#pragma unroll
    for (int j = 0; j < 8; ++j)
      rsum[j] += __shfl_xor(rsum[j], off, 32);
  if (l16 == 0) {
#pragma unroll
    for (int j = 0; j < 8; ++j) redbuf[(j + h8) * 8 + wave] = rsum[j];
  }
  __syncthreads();
  float inv[8];
#pragma unroll
  for (int j = 0; j < 8; ++j) {
    float s = 0.0f;
#pragma unroll
    for (int w = 0; w < 8; ++w) s += redbuf[(j + h8) * 8 + w];
    float qm = qmask[qrowbase + j + h8];
    inv[j] = qm / s;
  }

  float* outp = attnOut + ((long)hb << 20);
#pragma unroll
  for (int nb = 0; nb < 8; ++nb) {
    int col = nb * 128 + wave * 16 + l16;
#pragma unroll
    for (int j = 0; j < 8; ++j)
      outp[(q0 + j + h8) * SS + col] = acc[nb][j] * inv[j];
  }
}

// ------ final: concat([dec, ctx]) @ Wf + bf + dec  (K = 2048) -----------
__global__ __launch_bounds__(256) void k_gemm_final(const float* __restrict__ dec,
                                                    const float* __restrict__ ctx,
                                                    const float* __restrict__ Wf,
                                                    const float* __restrict__ bf,
                                                    float* __restrict__ xout)
{
  __shared__ float As[64 * 36];
  __shared__ float Bs[32 * 132];
  const int tid = threadIdx.x, lane = tid & 31, wave = tid >> 5;
  const int wm = wave & 3, wn = wave >> 2;
  const int l16 = lane & 15;
  const int kh2 = (lane >> 4) << 1;
  const int h8  = (lane >> 4) << 3;
  const int AS = 36, BS = 132;
  long m0 = (long)blockIdx.y * 64;
  long n0 = (long)blockIdx.x * 128;

  v8f acc[4];
#pragma unroll
  for (int i = 0; i < 4; ++i)
#pragma unroll
    for (int j = 0; j < 8; ++j) acc[i][j] = 0.0f;

  for (int k0 = 0; k0 < 2 * DD; k0 += 32) {
    const float* Asrc = (k0 < DD) ? (dec + m0 * DD + k0) : (ctx + m0 * DD + (k0 - DD));
#pragma unroll
    for (int i = 0; i < 2; ++i) {
      int id = tid + i * 256;
      int r = id >> 3, c = (id & 7) << 2;
      v4f v = *(const v4f*)(Asrc + (long)r * DD + c);
      *(v4f*)(As + r * AS + c) = v;
    }
#pragma unroll
    for (int i = 0; i < 4; ++i) {
      int id = tid + i * 256;
      int r = id >> 5, c = (id & 31) << 2;
      v4f v = *(const v4f*)(Wf + (long)(k0 + r) * DD + n0 + c);
      *(v4f*)(Bs + r * BS + c) = v;
      if (k0 + 32 < 2 * DD)
        __builtin_prefetch(Wf + (long)(k0 + 32 + r) * DD + n0 + c, 0, 1);
    }
    __syncthreads();
#pragma unroll
    for (int kk = 0; kk < 32; kk += 4) {
      v2f a;
      a.x = As[(wm * 16 + l16) * AS + kk + kh2];
      a.y = As[(wm * 16 + l16) * AS + kk + kh2 + 1];
#pragma unroll
      for (int nt = 0; nt < 4; ++nt) {
        v2f b;
        b.x = Bs[(kk + kh2) * BS + wn * 64 + nt * 16 + l16];
        b.y = Bs[(kk + kh2 + 1) * BS + wn * 64 + nt * 16 + l16];
        acc[nt] = wmma4(a, b, acc[nt]);
      }
    }
    __syncthreads();
  }
#pragma unroll
  for (int nt = 0; nt < 4; ++nt)
#pragma unroll
    for (int j = 0; j < 8; ++j) {
      long r = m0 + wm * 16 + j + h8;
      long c = n0 + wn * 64 + nt * 16 + l16;
      xout[r * DD + c] = acc[nt][j] + bf[c] + dec[r * DD + c];
    }
}

// ----------------------------- LayerNorm --------------------------------
__device__ __forceinline__ float blk_sum(float v, float* red) {
  int t = threadIdx.x;
  __syncthreads();
  red[t] = v;
  __syncthreads();
  for (int off = 128; off > 0; off >>= 1) {
    if (t < off) red[t] += red[t + off];
    __syncthreads();
  }
  return red[0];
}

__global__ __launch_bounds__(256) void k_layernorm(const float* __restrict__ X,
                                                   const float* __restrict__ gamma,
                                                   const float* __restrict__ beta,
                                                   float* __restrict__ out)
{
  __shared__ float red[256];
  long row = blockIdx.x;
  int t = threadIdx.x;
  float x[4];
  float s = 0.0f;
#pragma unroll
  for (int j = 0; j < 4; ++j) { x[j] = X[row * DD + t + j * 256]; s += x[j]; }
  float mu = blk_sum(s, red) * (1.0f / 1024.0f);
  float vs = 0.0f;
#pragma unroll
  for (int j = 0; j < 4; ++j) { float d = x[j] - mu; vs += d * d; }
  float var = blk_sum(vs, red) * (1.0f / 1024.0f);
  float rs = rsqrtf(var + 1e-5f);
#pragma unroll
  for (int j = 0; j < 4; ++j) {
    int c = t + j * 256;
    out[row * DD + c] = (x[j] - mu) * rs * gamma[c] + beta[c];
  }
}

// ---------------------------------------------------------------------------
extern "C" void kernel_launch(void* const* d_in, const int* in_sizes, int n_in,
                              void* d_out, int out_size, void* d_ws, size_t ws_size,
                              hipStream_t stream) {
  const float* memory = (const float*)d_in[0];
  const float* dec    = (const float*)d_in[1];
  const float* qmask  = (const float*)d_in[2];
  const float* Wk     = (const float*)d_in[3];
  const float* Wv     = (const float*)d_in[4];
  const float* Wq     = (const float*)d_in[5];
  const float* Wf     = (const float*)d_in[6];
  const float* bf     = (const float*)d_in[7];
  const float* gamma  = (const float*)d_in[8];
  const float* beta   = (const float*)d_in[9];
  const unsigned char* mask = (const unsigned char*)d_in[10];

  float* out = (float*)d_out;
  const size_t MD = (size_t)BB * SS * DD;       // 8,388,608
  float* attnOut = out + MD;                    // attns follow x in d_out
  float* ws = (float*)d_ws;
  float* Qb  = ws;                               // reused as xtmp later
  float* Kb  = ws + MD;                          // reused as ctx later
  float* Vb  = ws + 2 * MD;
  float* ctx = Kb;                               // Kb dead after k_attn
  float* xt  = Qb;                               // Qb dead after k_attn

  dim3 blk(256);
  k_gemm_proj<<<dim3(8, 128, 1), blk, 0, stream>>>(memory, Wk, Kb);
  k_gemm_proj<<<dim3(8, 128, 1), blk, 0, stream>>>(memory, Wv, Vb);
  k_gemm_proj<<<dim3(8, 128, 1), blk, 0, stream>>>(dec,    Wq, Qb);
  k_attn<<<dim3(64, 64, 1), blk, 0, stream>>>(Qb, Kb, mask, qmask, attnOut);
  k_gemm_av<<<dim3(1, 16, 64), blk, 0, stream>>>(attnOut, Vb, ctx);
  k_gemm_final<<<dim3(8, 128, 1), blk, 0, stream>>>(dec, ctx, Wf, bf, xt);
  k_layernorm<<<dim3(8192, 1, 1), blk, 0, stream>>>(xt, gamma, beta, out);
}